// self_training_conditional_72825465471291
// MI455X (gfx1250) — compile-verified
//
#include <hip/hip_runtime.h>
#include <math.h>

typedef __attribute__((ext_vector_type(2))) float v2f;
typedef __attribute__((ext_vector_type(8))) float v8f;

#define NT 8192
#define NS 8192
#define CDIM 512
#define THRESH 0.01f

// ---------------------------------------------------------------------------
// Kernel 1: per-row max / argmax (first-index tie-break, like jnp.argmax).
// One wave32 per row; 8 rows per 256-thread block.
// Optionally writes float label/loss (target pass); target mask/label go straight into d_out.
// ---------------------------------------------------------------------------
__global__ void row_stats_kernel(const float* __restrict__ mat,
                                 int*   __restrict__ lab_i,
                                 float* __restrict__ mask_f,
                                 float* __restrict__ lab_f,   // nullable
                                 float* __restrict__ loss)    // nullable
{
    const int wave = threadIdx.x >> 5;
    const int lane = threadIdx.x & 31;
    const int row  = blockIdx.x * (blockDim.x >> 5) + wave;

    const float* r = mat + (size_t)row * CDIM;
    float best = -1.0f;
    int   bidx = 0;
    for (int k = lane; k < CDIM; k += 32) {
        float v = r[k];
        if (v > best) { best = v; bidx = k; }   // strict > keeps first index
    }
    // wave32 reduction: larger value wins; tie -> smaller index
    for (int off = 16; off >= 1; off >>= 1) {
        float ov = __shfl_xor(best, off, 32);
        int   oi = __shfl_xor(bidx, off, 32);
        if (ov > best || (ov == best && oi < bidx)) { best = ov; bidx = oi; }
    }
    if (lane == 0) {
        lab_i[row]  = bidx;
        mask_f[row] = (best > THRESH) ? 1.0f : 0.0f;
        if (lab_f) lab_f[row] = (float)bidx;
        if (loss)  loss[row]  = -logf(best);   // -log(td[i, argmax]) == -log(max)
    }
}

// ---------------------------------------------------------------------------
// Kernel 2: deterministic class sums  S[c,:] = sum of masked sd rows of class c.
// One block per class; labels (masked) staged in LDS; fixed j order -> no
// float atomics, fully deterministic. Each thread owns 2 of the 512 columns.
// ---------------------------------------------------------------------------
__global__ void class_sum_kernel(const float* __restrict__ sd,
                                 const int*   __restrict__ lab_s,
                                 const float* __restrict__ mask_s,
                                 float* __restrict__ S)
{
    __shared__ int slab[NS];          // 32 KB of the 320 KB WGP LDS
    const int c = blockIdx.x;
    const int t = threadIdx.x;        // 0..255

    for (int j = t; j < NS; j += 256)
        slab[j] = (mask_s[j] != 0.0f) ? lab_s[j] : -1;   // -1 never matches
    __syncthreads();

    float acc0 = 0.0f, acc1 = 0.0f;
    for (int j = 0; j < NS; ++j) {
        if (slab[j] == c) {
            const float* r = sd + (size_t)j * CDIM;
            acc0 += r[t];
            acc1 += r[t + 256];
        }
    }
    S[(size_t)c * CDIM + t]       = acc0;
    S[(size_t)c * CDIM + t + 256] = acc1;
}

// ---------------------------------------------------------------------------
// Kernel 3: WMMA score. One wave32 per 16-row tile:
//   D(16x16) = A(16x512) x B(512x16), col n of B = S[lab_t[base+n], :]
//   score[base+m] = D[m][m]
// Uses V_WMMA_F32_16X16X4_F32 (fp32 in/out — exact for fp32 inputs),
// 128 chained WMMAs over K=512.
// ISA f32 layouts: A 16x4: V0 = K0(lanes 0-15)/K2(lanes 16-31), V1 = K1/K3,
// M = lane&15. B 4x16 mirrors it with N = lane&15 -> identical per-lane
// addressing for both operands.
// ---------------------------------------------------------------------------
__global__ void wmma_score_kernel(const float* __restrict__ td,
                                  const int*   __restrict__ lab_t,
                                  const float* __restrict__ S,
                                  float* __restrict__ score)
{
    const int lane = threadIdx.x & 31;
    const int base = blockIdx.x * 16;
    const int m    = lane & 15;
    const int kSel = (lane >> 4) << 1;   // 0 for lanes 0-15, 2 for lanes 16-31

    const float* aRow = td + (size_t)(base + m) * CDIM;
    const float* bRow = S  + (size_t)lab_t[base + m] * CDIM;

    v8f acc = {};
    for (int kk = 0; kk < CDIM; kk += 4) {
        v2f a = *(const v2f*)(aRow + kk + kSel);   // 8B aligned (even offsets)
        v2f b = *(const v2f*)(bRow + kk + kSel);
        acc = __builtin_amdgcn_wmma_f32_16x16x4_f32(
            /*neg_a=*/false, a, /*neg_b=*/false, b,
            /*c_mod=*/(short)0, acc, /*reuse_a=*/false, /*reuse_b=*/false);
    }

    float a8[8];
#pragma unroll
    for (int i = 0; i < 8; ++i) a8[i] = acc[i];

    // diagonal of 32-bit 16x16 C/D: lane l<8 holds D[l][l] in VGPR l;
    // lane l in 24..31 holds D[l-16][l-16] in VGPR l-24.
    if (lane < 8)        score[base + lane]      = a8[lane];
    else if (lane >= 24) score[base + lane - 16] = a8[lane - 24];
}

// ---------------------------------------------------------------------------
// Kernel 4: masked softmax over score + weighted loss. Single 1024-thread
// block, fixed-order tree reductions (deterministic). Reproduces
// softmax(where(mask, score, -1e30)) exactly, including the all-masked case.
// ---------------------------------------------------------------------------
__global__ void finalize_kernel(const float* __restrict__ score,
                                const float* __restrict__ mask_f,
                                const float* __restrict__ loss,
                                float* __restrict__ weights,
                                float* __restrict__ out0)
{
    __shared__ float red[1024];
    const int t = threadIdx.x;

    float ms[8], mk[8];
    float lmax = -1e30f;
#pragma unroll
    for (int i = 0; i < 8; ++i) {
        int idx = t + i * 1024;
        float mki = mask_f[idx];
        float s   = (mki != 0.0f) ? score[idx] : -1e30f;
        ms[i] = s; mk[i] = mki;
        lmax = fmaxf(lmax, s);
    }
    red[t] = lmax; __syncthreads();
    for (int off = 512; off >= 1; off >>= 1) {
        if (t < off) red[t] = fmaxf(red[t], red[t + off]);
        __syncthreads();
    }
    const float gmax = red[0];
    __syncthreads();

    float e[8];
    float lsum = 0.0f, lcnt = 0.0f;
#pragma unroll
    for (int i = 0; i < 8; ++i) {
        e[i] = expf(ms[i] - gmax);   // underflows to 0 for masked-out rows
        lsum += e[i];
        lcnt += mk[i];
    }
    red[t] = lsum; __syncthreads();
    for (int off = 512; off >= 1; off >>= 1) {
        if (t < off) red[t] += red[t + off];
        __syncthreads();
    }
    const float denom = red[0];
    __syncthreads();
    red[t] = lcnt; __syncthreads();
    for (int off = 512; off >= 1; off >>= 1) {
        if (t < off) red[t] += red[t + off];
        __syncthreads();
    }
    const float cnt = red[0];
    __syncthreads();

    float wacc = 0.0f;
#pragma unroll
    for (int i = 0; i < 8; ++i) {
        int idx = t + i * 1024;
        float w = e[i] / denom;
        weights[idx] = w;
        wacc += mk[i] * loss[idx] * w;
    }
    red[t] = wacc; __syncthreads();
    for (int off = 512; off >= 1; off >>= 1) {
        if (t < off) red[t] += red[t + off];
        __syncthreads();
    }
    if (t == 0) out0[0] = red[0] / fmaxf(cnt, 1.0f);
}

// ---------------------------------------------------------------------------
extern "C" void kernel_launch(void* const* d_in, const int* in_sizes, int n_in,
                              void* d_out, int out_size, void* d_ws, size_t ws_size,
                              hipStream_t stream) {
    const float* td = (const float*)d_in[0];   // [NT, C] target probs
    const float* sd = (const float*)d_in[1];   // [NS, C] source probs
    float* out = (float*)d_out;                // [1 + NT + NT + NT]

    char* ws = (char*)d_ws;
    float* S      = (float*)ws;                                 // 512*512 f32 (1 MB)
    int*   lab_s  = (int*)  (ws + (size_t)CDIM * CDIM * 4);     // 8192 i32
    float* mask_s = (float*)(ws + (size_t)CDIM * CDIM * 4 + 32768);
    int*   lab_ti = (int*)  (ws + (size_t)CDIM * CDIM * 4 + 65536);
    float* loss   = (float*)(ws + (size_t)CDIM * CDIM * 4 + 98304);
    float* score  = (float*)(ws + (size_t)CDIM * CDIM * 4 + 131072);

    float* mask_out = out + 1;            // mask_t as 0/1 floats
    float* lab_out  = out + 1 + NT;       // lab_t as floats
    float* w_out    = out + 1 + 2 * NT;   // weights

    // 1) source row stats (max/argmax/mask)
    row_stats_kernel<<<NS / 8, 256, 0, stream>>>(sd, lab_s, mask_s, nullptr, nullptr);
    // 2) target row stats + per-row NLL loss, masks/labels straight to d_out
    row_stats_kernel<<<NT / 8, 256, 0, stream>>>(td, lab_ti, mask_out, lab_out, loss);
    // 3) deterministic class sums S[c,:]
    class_sum_kernel<<<CDIM, 256, 0, stream>>>(sd, lab_s, mask_s, S);
    // 4) WMMA gathered-GEMM diagonal -> score (512 tiles, one wave32 each)
    wmma_score_kernel<<<NT / 16, 32, 0, stream>>>(td, lab_ti, S, score);
    // 5) masked softmax + weighted loss
    finalize_kernel<<<1, 1024, 0, stream>>>(score, mask_out, loss, w_out, out);
}